// ILRMA_V2_63866163692382
// MI455X (gfx1250) — compile-verified
//
#include <hip/hip_runtime.h>

// ---------------- problem constants (from reference setup_inputs) ----------
#define I_DIM 1025
#define I_PAD 1040            // 65*16, zero-padded rows for 16-row WMMA tiles
#define J_DIM 1024
#define K_DIM 8
#define NITER 10
#define EPSF  1e-20f
#define REGF  1e-6f

typedef float v2f __attribute__((ext_vector_type(2)));
typedef float v8f __attribute__((ext_vector_type(8)));

// FP32 WMMA 16x16x4: D = A(16x4) * B(4x16) + C(16x16), exact fp32 math.
__device__ __forceinline__ v8f wmma4(v2f a, v2f b, v8f c) {
  return __builtin_amdgcn_wmma_f32_16x16x4_f32(false, a, false, b, (short)0, c,
                                               false, false);
}
// Hardware v_rcp_f32 (~1 ULP) for hot per-element reciprocals.
__device__ __forceinline__ float frcp(float x) {
  return __builtin_amdgcn_rcpf(x);
}

// --------------------------------------------------------------------------
// Pack X [M,J,I,2] -> Xc [I_PAD][M][J][2] (pad rows zero), and init
// Y2[n][I_PAD][J] = |X|^2 (W starts as identity, so initial Y == X_c).
__global__ void k_pack_x(const float* __restrict__ X, float* __restrict__ Xc,
                         float* __restrict__ Y2) {
  int t = blockIdx.x * blockDim.x + threadIdx.x;
  if (t >= I_PAD * J_DIM) return;
  int i = t / J_DIM, j = t - i * J_DIM;
#pragma unroll
  for (int m = 0; m < 2; ++m) {
    float re = 0.f, im = 0.f;
    if (i < I_DIM) {
      int src = ((m * J_DIM + j) * I_DIM + i) * 2;
      re = X[src]; im = X[src + 1];
    }
    int dst = ((i * 2 + m) * J_DIM + j) * 2;
    Xc[dst] = re; Xc[dst + 1] = im;
    Y2[m * (I_PAD * J_DIM) + t] = re * re + im * im;
  }
}

// Pack T0 [I,K,M] -> T[n][i_pad][k] (pad rows = 0), V0 [K,J,M] -> V[n][k][j],
// W[i][2][2] complex = identity.
__global__ void k_pack_tvw(const float* __restrict__ T0,
                           const float* __restrict__ V0,
                           float* __restrict__ T, float* __restrict__ V,
                           float* __restrict__ W) {
  int t = blockIdx.x * blockDim.x + threadIdx.x;
  const int nT = 2 * I_PAD * K_DIM;
  if (t < nT) {
    int n = t / (I_PAD * K_DIM);
    int r = t - n * (I_PAD * K_DIM);
    int i = r / K_DIM, k = r - i * K_DIM;
    T[t] = (i < I_DIM) ? T0[(i * K_DIM + k) * 2 + n] : 0.f;
    return;
  }
  t -= nT;
  const int nV = 2 * K_DIM * J_DIM;
  if (t < nV) {
    int n = t / (K_DIM * J_DIM);
    int r = t - n * (K_DIM * J_DIM);
    V[t] = V0[r * 2 + n];
    return;
  }
  t -= nV;
  if (t < I_DIM * 8) {
    int c = t & 7;                          // [row][col][re/im]
    W[t] = (c == 0 || c == 6) ? 1.f : 0.f;  // identity
  }
}

// --------------------------------------------------------------------------
// K1: T update for source n. One block = one 16-row i-tile, 4 waves split J.
// Per 16-j step:  R tile  = Told(16x8) @ V(8x16)        -> 2  wmma
//                 numT += Yd(16x16) @ V^T(16x8-in-16)   -> 4  wmma
//                 denT += Rd(16x16) @ V^T                -> 4  wmma
__global__ __launch_bounds__(128) void k1_tupd(const float* __restrict__ Told,
                                               float* __restrict__ Tnew,
                                               const float* __restrict__ Vn,
                                               const float* __restrict__ Y2n) {
  __shared__ float sV[K_DIM * J_DIM];        // 32 KB: whole V panel
  __shared__ float sYd[4][16 * 17];          // per-wave C->A transpose tiles
  __shared__ float sRd[4][16 * 17];
  __shared__ float rN[16 * 16];              // cross-wave accumulators
  __shared__ float rD[16 * 16];

  const int tid = threadIdx.x;
  const int wave = tid >> 5, lane = tid & 31;
  const int hi = lane >> 4, lo = lane & 15;
  const int i0 = blockIdx.x * 16;

  for (int idx = tid; idx < K_DIM * J_DIM; idx += 128) sV[idx] = Vn[idx];
  for (int idx = tid; idx < 256; idx += 128) { rN[idx] = 0.f; rD[idx] = 0.f; }
  __syncthreads();

  // A tiles of Told (constant over J): rows i0+lo, K = 2*hi + v (+4 for aT2)
  const int tb = (i0 + lo) * K_DIM;
  v2f aT1, aT2;
  aT1.x = Told[tb + 2 * hi];     aT1.y = Told[tb + 2 * hi + 1];
  aT2.x = Told[tb + 4 + 2 * hi]; aT2.y = Told[tb + 4 + 2 * hi + 1];

  v8f accN = {0.f, 0.f, 0.f, 0.f, 0.f, 0.f, 0.f, 0.f};
  v8f accD = {0.f, 0.f, 0.f, 0.f, 0.f, 0.f, 0.f, 0.f};
  float* myYd = sYd[wave];
  float* myRd = sRd[wave];

  for (int step = wave; step < J_DIM / 16; step += 4) {
    const int j0 = step * 16;
    v2f b1, b2;
    b1.x = sV[(2 * hi) * J_DIM + j0 + lo];
    b1.y = sV[(2 * hi + 1) * J_DIM + j0 + lo];
    b2.x = sV[(4 + 2 * hi) * J_DIM + j0 + lo];
    b2.y = sV[(5 + 2 * hi) * J_DIM + j0 + lo];
    v8f accR = {0.f, 0.f, 0.f, 0.f, 0.f, 0.f, 0.f, 0.f};
    accR = wmma4(aT1, b1, accR);
    accR = wmma4(aT2, b2, accR);
#pragma unroll
    for (int v = 0; v < 8; ++v) {      // C layout: row = v+8*hi, col = lo
      int row = v + 8 * hi;
      int gi = i0 + row;               // < I_PAD, Y2 padded -> no guard
      float r = accR[v];
      float y2 = Y2n[gi * J_DIM + j0 + lo];
      myYd[row * 17 + lo] = y2 * frcp(r * r + EPSF);
      myRd[row * 17 + lo] = frcp(r + EPSF);
    }
#pragma unroll
    for (int c = 0; c < 4; ++c) {      // K-chunks of 4 j's
      const int acol = 4 * c + 2 * hi;
      v2f aY, aR, bT;
      aY.x = myYd[lo * 17 + acol];  aY.y = myYd[lo * 17 + acol + 1];
      aR.x = myRd[lo * 17 + acol];  aR.y = myRd[lo * 17 + acol + 1];
      bT.x = (lo < K_DIM) ? sV[lo * J_DIM + j0 + acol] : 0.f;
      bT.y = (lo < K_DIM) ? sV[lo * J_DIM + j0 + acol + 1] : 0.f;
      accN = wmma4(aY, bT, accN);
      accD = wmma4(aR, bT, accD);
    }
  }
#pragma unroll
  for (int v = 0; v < 8; ++v) {
    int row = v + 8 * hi;
    atomicAdd(&rN[row * 16 + lo], accN[v]);
    atomicAdd(&rD[row * 16 + lo], accD[v]);
  }
  __syncthreads();
  if (wave == 0 && lo < K_DIM) {
#pragma unroll
    for (int v = 0; v < 8; ++v) {
      int row = v + 8 * hi;
      int gi = i0 + row;
      float num = rN[row * 16 + lo];
      float den = rD[row * 16 + lo];
      float told = Told[gi * K_DIM + lo];
      Tnew[gi * K_DIM + lo] = told * sqrtf(num / (den + EPSF));
    }
  }
}

// --------------------------------------------------------------------------
// K2: V update. One thread per frame j; reduce over all I bins.
// Uses OLD T (for R/Yd/Rd) and NEW T (for num/den), per reference semantics.
__global__ __launch_bounds__(256) void k2_vupd(const float* __restrict__ Told,
                                               const float* __restrict__ Tnew,
                                               float* __restrict__ Vn,
                                               const float* __restrict__ Y2n) {
  int j = blockIdx.x * blockDim.x + threadIdx.x;
  if (j >= J_DIM) return;
  float vold[K_DIM], numv[K_DIM], denv[K_DIM];
#pragma unroll
  for (int k = 0; k < K_DIM; ++k) {
    vold[k] = Vn[k * J_DIM + j];
    numv[k] = 0.f; denv[k] = 0.f;
  }
  for (int i = 0; i < I_DIM; ++i) {
    const float* to = Told + i * K_DIM;
    const float* tn = Tnew + i * K_DIM;
    float r = 0.f;
#pragma unroll
    for (int k = 0; k < K_DIM; ++k) r += to[k] * vold[k];
    float y2 = Y2n[i * J_DIM + j];
    float yd = y2 * frcp(r * r + EPSF);
    float rd = frcp(r + EPSF);
#pragma unroll
    for (int k = 0; k < K_DIM; ++k) {
      float t = tn[k];
      numv[k] += t * yd;
      denv[k] += t * rd;
    }
  }
#pragma unroll
  for (int k = 0; k < K_DIM; ++k)
    Vn[k * J_DIM + j] = vold[k] * sqrtf(numv[k] / (denv[k] + EPSF));
}

// --------------------------------------------------------------------------
// K3: demixing update for source n. One block = 16 bins. R_new tiles via WMMA;
// weighted 2x2 Hermitian covariance accumulated per lane, reduced via LDS
// ds_add_f32, then a per-bin 2x2 complex solve updates W row n.
__global__ __launch_bounds__(128) void k3_wupd(const float* __restrict__ Tn,
                                               const float* __restrict__ Vn,
                                               const float* __restrict__ Xc,
                                               float* __restrict__ W, int n) {
  __shared__ float sV[K_DIM * J_DIM];     // 32 KB
  __shared__ float red[4 * 16 * 16];      // d00, d01r, d01i, d11

  const int tid = threadIdx.x;
  const int wave = tid >> 5, lane = tid & 31;
  const int hi = lane >> 4, lo = lane & 15;
  const int i0 = blockIdx.x * 16;

  for (int idx = tid; idx < K_DIM * J_DIM; idx += 128) sV[idx] = Vn[idx];
  for (int idx = tid; idx < 4 * 256; idx += 128) red[idx] = 0.f;
  __syncthreads();

  const int tb = (i0 + lo) * K_DIM;
  v2f aT1, aT2;
  aT1.x = Tn[tb + 2 * hi];     aT1.y = Tn[tb + 2 * hi + 1];
  aT2.x = Tn[tb + 4 + 2 * hi]; aT2.y = Tn[tb + 4 + 2 * hi + 1];

  float d00[8], d01r[8], d01i[8], d11[8];
#pragma unroll
  for (int v = 0; v < 8; ++v) { d00[v] = d01r[v] = d01i[v] = d11[v] = 0.f; }

  const float2* xp = (const float2*)Xc;   // Xc padded to I_PAD rows (zeros)
  for (int step = wave; step < J_DIM / 16; step += 4) {
    const int j0 = step * 16;
    v2f b1, b2;
    b1.x = sV[(2 * hi) * J_DIM + j0 + lo];
    b1.y = sV[(2 * hi + 1) * J_DIM + j0 + lo];
    b2.x = sV[(4 + 2 * hi) * J_DIM + j0 + lo];
    b2.y = sV[(5 + 2 * hi) * J_DIM + j0 + lo];
    v8f accR = {0.f, 0.f, 0.f, 0.f, 0.f, 0.f, 0.f, 0.f};
    accR = wmma4(aT1, b1, accR);
    accR = wmma4(aT2, b2, accR);
#pragma unroll
    for (int v = 0; v < 8; ++v) {
      int row = v + 8 * hi;
      int gi = i0 + row;
      int col = j0 + lo;
      float wgt = frcp(accR[v] + EPSF);
      float2 x0 = xp[(gi * 2) * J_DIM + col];
      float2 x1 = xp[(gi * 2 + 1) * J_DIM + col];
      d00[v]  += (x0.x * x0.x + x0.y * x0.y) * wgt;
      d11[v]  += (x1.x * x1.x + x1.y * x1.y) * wgt;
      d01r[v] += (x0.x * x1.x + x0.y * x1.y) * wgt;   // x0 * conj(x1)
      d01i[v] += (x0.y * x1.x - x0.x * x1.y) * wgt;
    }
  }
#pragma unroll
  for (int v = 0; v < 8; ++v) {
    int row = v + 8 * hi;
    atomicAdd(&red[0 * 256 + row * 16 + lo], d00[v]);
    atomicAdd(&red[1 * 256 + row * 16 + lo], d01r[v]);
    atomicAdd(&red[2 * 256 + row * 16 + lo], d01i[v]);
    atomicAdd(&red[3 * 256 + row * 16 + lo], d11[v]);
  }
  __syncthreads();

  if (tid < 16) {
    int gi = i0 + tid;
    if (gi < I_DIM) {
      float s00 = 0.f, s01r = 0.f, s01i = 0.f, s11 = 0.f;
#pragma unroll
      for (int t = 0; t < 16; ++t) {
        s00  += red[0 * 256 + tid * 16 + t];
        s01r += red[1 * 256 + tid * 16 + t];
        s01i += red[2 * 256 + tid * 16 + t];
        s11  += red[3 * 256 + tid * 16 + t];
      }
      const float invJ = 1.0f / (float)J_DIM;
      float D00 = s00 * invJ + REGF;
      float D01r = s01r * invJ, D01i = s01i * invJ;
      float D11 = s11 * invJ + REGF;
      const float* wp = W + gi * 8;
      float w00r = wp[0], w00i = wp[1], w01r = wp[2], w01i = wp[3];
      float w10r = wp[4], w10i = wp[5], w11r = wp[6], w11i = wp[7];
      // A = W @ D,  D = [[D00, D01], [conj(D01), D11]]
      float a00r = w00r * D00 + w01r * D01r + w01i * D01i;
      float a00i = w00i * D00 + w01i * D01r - w01r * D01i;
      float a01r = w00r * D01r - w00i * D01i + w01r * D11;
      float a01i = w00r * D01i + w00i * D01r + w01i * D11;
      float a10r = w10r * D00 + w11r * D01r + w11i * D01i;
      float a10i = w10i * D00 + w11i * D01r - w11r * D01i;
      float a11r = w10r * D01r - w10i * D01i + w11r * D11;
      float a11i = w10r * D01i + w10i * D01r + w11i * D11;
      float detr = a00r * a11r - a00i * a11i - a01r * a10r + a01i * a10i;
      float deti = a00r * a11i + a00i * a11r - a01r * a10i - a01i * a10r;
      float u0r, u0i, u1r, u1i;               // column n of adj(A)
      if (n == 0) { u0r =  a11r; u0i =  a11i; u1r = -a10r; u1i = -a10i; }
      else        { u0r = -a01r; u0i = -a01i; u1r =  a00r; u1i =  a00i; }
      float idet = 1.0f / (detr * detr + deti * deti);
      float b0r = (u0r * detr + u0i * deti) * idet;
      float b0i = (u0i * detr - u0r * deti) * idet;
      float b1r = (u1r * detr + u1i * deti) * idet;
      float b1i = (u1i * detr - u1r * deti) * idet;
      float cre = b0r * b1r + b0i * b1i;       // conj(b0)*b1
      float cim = b0r * b1i - b0i * b1r;
      float q = D00 * (b0r * b0r + b0i * b0i) + D11 * (b1r * b1r + b1i * b1i) +
                2.0f * (D01r * cre - D01i * cim);
      float dn = sqrtf(q + EPSF);
      float* wo = W + gi * 8 + n * 4;          // W[gi][n][:] = conj(b)/dn
      wo[0] =  b0r / dn; wo[1] = -b0i / dn;
      wo[2] =  b1r / dn; wo[3] = -b1i / dn;
    }
  }
}

// --------------------------------------------------------------------------
// K4: Y2[n][i][j] = |(W @ X_c)[i][n][j]|^2 for next iteration's NMF.
// Only valid rows are written; Y2 pad rows stay zero.
__global__ void k4_y2(const float* __restrict__ W, const float* __restrict__ Xc,
                      float* __restrict__ Y2) {
  int t = blockIdx.x * blockDim.x + threadIdx.x;
  if (t >= I_DIM * J_DIM) return;
  int i = t / J_DIM, j = t - i * J_DIM;
  const float2* xp = (const float2*)Xc;
  float2 x0 = xp[(i * 2) * J_DIM + j];
  float2 x1 = xp[(i * 2 + 1) * J_DIM + j];
#pragma unroll
  for (int n = 0; n < 2; ++n) {
    float wr0 = W[i * 8 + n * 4 + 0], wi0 = W[i * 8 + n * 4 + 1];
    float wr1 = W[i * 8 + n * 4 + 2], wi1 = W[i * 8 + n * 4 + 3];
    float yr = wr0 * x0.x - wi0 * x0.y + wr1 * x1.x - wi1 * x1.y;
    float yi = wr0 * x0.y + wi0 * x0.x + wr1 * x1.y + wi1 * x1.x;
    Y2[n * (I_PAD * J_DIM) + i * J_DIM + j] = yr * yr + yi * yi;
  }
}

// K5: final Y = W @ X_c, emitted as [N, J, I, 2] (coalesced over i).
__global__ void k5_out(const float* __restrict__ W, const float* __restrict__ Xc,
                       float* __restrict__ out) {
  int t = blockIdx.x * blockDim.x + threadIdx.x;
  if (t >= I_DIM * J_DIM) return;
  int i = t % I_DIM, j = t / I_DIM;
  const float2* xp = (const float2*)Xc;
  float2 x0 = xp[(i * 2) * J_DIM + j];
  float2 x1 = xp[(i * 2 + 1) * J_DIM + j];
  float2* op = (float2*)out;
#pragma unroll
  for (int n = 0; n < 2; ++n) {
    float wr0 = W[i * 8 + n * 4 + 0], wi0 = W[i * 8 + n * 4 + 1];
    float wr1 = W[i * 8 + n * 4 + 2], wi1 = W[i * 8 + n * 4 + 3];
    float2 y;
    y.x = wr0 * x0.x - wi0 * x0.y + wr1 * x1.x - wi1 * x1.y;
    y.y = wr0 * x0.y + wi0 * x0.x + wr1 * x1.y + wi1 * x1.x;
    op[(n * J_DIM + j) * I_DIM + i] = y;
  }
}

// --------------------------------------------------------------------------
extern "C" void kernel_launch(void* const* d_in, const int* in_sizes, int n_in,
                              void* d_out, int out_size, void* d_ws,
                              size_t ws_size, hipStream_t stream) {
  const float* X  = (const float*)d_in[0];   // [M,J,I,2]
  const float* T0 = (const float*)d_in[1];   // [I,K,M]
  const float* V0 = (const float*)d_in[2];   // [K,J,M]
  float* out = (float*)d_out;
  float* ws = (float*)d_ws;

  const int IJ  = I_DIM * J_DIM;
  const int IPJ = I_PAD * J_DIM;

  size_t off = 0;
  float* Xc = ws + off; off += (size_t)I_PAD * 2 * J_DIM * 2;   // 17.0 MB
  float* Y2 = ws + off; off += (size_t)2 * IPJ;                 //  8.5 MB
  float* Ta = ws + off; off += (size_t)2 * I_PAD * K_DIM;
  float* Tb = ws + off; off += (size_t)2 * I_PAD * K_DIM;
  float* V  = ws + off; off += (size_t)2 * K_DIM * J_DIM;
  float* W  = ws + off; off += (size_t)I_DIM * 8;

  k_pack_x<<<(IPJ + 255) / 256, 256, 0, stream>>>(X, Xc, Y2);
  {
    int tot = 2 * I_PAD * K_DIM + 2 * K_DIM * J_DIM + I_DIM * 8;
    k_pack_tvw<<<(tot + 255) / 256, 256, 0, stream>>>(T0, V0, Ta, V, W);
  }

  float* Tcur[2] = {Ta, Ta + I_PAD * K_DIM};
  float* Talt[2] = {Tb, Tb + I_PAD * K_DIM};

  for (int it = 0; it < NITER; ++it) {
    for (int n = 0; n < 2; ++n) {
      float* Vn  = V + n * K_DIM * J_DIM;
      float* Y2n = Y2 + (size_t)n * IPJ;
      k1_tupd<<<65, 128, 0, stream>>>(Tcur[n], Talt[n], Vn, Y2n);
      k2_vupd<<<J_DIM / 256, 256, 0, stream>>>(Tcur[n], Talt[n], Vn, Y2n);
      k3_wupd<<<65, 128, 0, stream>>>(Talt[n], Vn, Xc, W, n);
      float* tmp = Tcur[n]; Tcur[n] = Talt[n]; Talt[n] = tmp;  // ping-pong
    }
    k4_y2<<<(IJ + 255) / 256, 256, 0, stream>>>(W, Xc, Y2);
  }
  k5_out<<<(IJ + 255) / 256, 256, 0, stream>>>(W, Xc, out);
}